// LightGCN_29154238005727
// MI455X (gfx1250) — compile-verified
//
#include <hip/hip_runtime.h>
#include <hip/hip_bf16.h>

#define N_USERS 100000
#define N_ITEMS 50000
#define EMB_DIM 64
#define N_LAYERS 3
#define BATCH   16384
#define N_NODES (N_USERS + N_ITEMS)

typedef __attribute__((ext_vector_type(2))) float v2f;
typedef __attribute__((ext_vector_type(8))) float v8f;

// ---------------------------------------------------------------- utilities
__global__ void zero_f32(float* __restrict__ p, size_t n) {
    size_t i = (size_t)blockIdx.x * blockDim.x + threadIdx.x;
    if (i < n) p[i] = 0.0f;
}

// deg[row] += 1 for both directions of every bipartite edge
__global__ void deg_kernel(const int* __restrict__ eu, const int* __restrict__ ei,
                           float* __restrict__ deg, int nedges) {
    int e = blockIdx.x * blockDim.x + threadIdx.x;
    if (e >= nedges) return;
    atomicAdd(&deg[eu[e]], 1.0f);
    atomicAdd(&deg[N_USERS + ei[e]], 1.0f);
}

// deg -> (deg + 1e-8)^-1/2 in place
__global__ void dinv_kernel(float* __restrict__ deg, int n) {
    int i = blockIdx.x * blockDim.x + threadIdx.x;
    if (i < n) deg[i] = 1.0f / sqrtf(deg[i] + 1e-8f);
}

// x0 = acc = concat(user_emb, item_emb)
__global__ void init_emb(const float* __restrict__ ue, const float* __restrict__ ie,
                         float* __restrict__ x, float* __restrict__ acc) {
    size_t i = (size_t)blockIdx.x * blockDim.x + threadIdx.x;
    size_t total = (size_t)N_NODES * EMB_DIM;
    if (i >= total) return;
    size_t ucut = (size_t)N_USERS * EMB_DIM;
    float v = (i < ucut) ? ue[i] : ie[i - ucut];
    x[i] = v;
    acc[i] = v;
}

// One wave32 per edge; lanes cover the 64 dims as float2.
// xout[u]  += val * xin[it];  xout[it] += val * xin[u]
__global__ void scatter_kernel(const int* __restrict__ eu, const int* __restrict__ ei,
                               const float* __restrict__ dinv,
                               const float* __restrict__ xin,
                               float* __restrict__ xout, int nedges) {
    int t    = blockIdx.x * blockDim.x + threadIdx.x;
    int e    = t >> 5;
    int lane = t & 31;
    if (e >= nedges) return;
    int u  = eu[e];
    int it = N_USERS + ei[e];
    float val = dinv[u] * dinv[it];
    const float2* xu = (const float2*)(xin + (size_t)u  * EMB_DIM);
    const float2* xi = (const float2*)(xin + (size_t)it * EMB_DIM);
    float2 a = xu[lane];
    float2 b = xi[lane];
    float* ou = xout + (size_t)u  * EMB_DIM + lane * 2;
    float* oi = xout + (size_t)it * EMB_DIM + lane * 2;
    atomicAdd(ou,     val * b.x);
    atomicAdd(ou + 1, val * b.y);
    atomicAdd(oi,     val * a.x);
    atomicAdd(oi + 1, val * a.y);
}

__global__ void acc_add(float* __restrict__ acc, const float* __restrict__ x, size_t n) {
    size_t i = (size_t)blockIdx.x * blockDim.x + threadIdx.x;
    if (i < n) acc[i] += x[i];
}

// ------------------------------------------------- final batched dot via WMMA
// One wave per 16 (user,item) pairs. A = UE tile (16x64), B = IE tile^T
// (64x16); accumulate 16x V_WMMA_F32_16X16X4_F32 (fp32, K=4 each), read the
// diagonal of D. final = acc/4 on both sides -> scale dot by 1/16.
__global__ void final_dot_wmma(const float* __restrict__ acc,
                               const int* __restrict__ user_ids,
                               const int* __restrict__ item_ids,
                               const float* __restrict__ user_bias,
                               const float* __restrict__ item_bias,
                               float* __restrict__ out) {
    const int lane = threadIdx.x & 31;
    const int wave = threadIdx.x >> 5;
    const int tile = blockIdx.x * (blockDim.x >> 5) + wave;
    if (tile >= BATCH / 16) return;           // uniform per wave

    const int m = lane & 15;                  // A row == B col == batch slot
    const int g = lane >> 4;                  // lane half selects K pair
    const int b = tile * 16 + m;
    const int uid = user_ids[b];
    const int iid = item_ids[b];
    const float* rowU = acc + (size_t)uid * EMB_DIM;
    const float* rowI = acc + ((size_t)N_USERS + (size_t)iid) * EMB_DIM;

    v8f c = {0.f, 0.f, 0.f, 0.f, 0.f, 0.f, 0.f, 0.f};
#pragma unroll
    for (int kk = 0; kk < EMB_DIM; kk += 4) {
        // A 16x4 f32: lane(m,g), vgpr v -> K = 2g+v  => consecutive pair
        v2f a  = *(const v2f*)(rowU + kk + 2 * g);
        v2f bb = *(const v2f*)(rowI + kk + 2 * g);
        c = __builtin_amdgcn_wmma_f32_16x16x4_f32(
                /*neg_a=*/false, a, /*neg_b=*/false, bb,
                /*c_mod=*/(short)0, c, /*reuse_a=*/false, /*reuse_b=*/false);
    }

    // D 16x16 f32: vgpr r, lane L -> M = r + 8*(L/16), N = L%16.
    // Diagonal (M==N): lanes 0..7 -> c[lane]; lanes 24..31 -> c[lane-24].
    int r = (g == 0) ? ((m < 8) ? m : -1) : ((m >= 8) ? (m - 8) : -1);
    if (r >= 0) {
        float d;
        switch (r) {
            case 0: d = c[0]; break; case 1: d = c[1]; break;
            case 2: d = c[2]; break; case 3: d = c[3]; break;
            case 4: d = c[4]; break; case 5: d = c[5]; break;
            case 6: d = c[6]; break; default: d = c[7]; break;
        }
        out[b] = d * 0.0625f + user_bias[uid] + item_bias[iid]; // + GLOBAL_MEAN(0)
    }
}

// ---------------------------------------------------------------- launcher
extern "C" void kernel_launch(void* const* d_in, const int* in_sizes, int n_in,
                              void* d_out, int out_size, void* d_ws, size_t ws_size,
                              hipStream_t stream) {
    const float* user_emb  = (const float*)d_in[0];
    const float* item_emb  = (const float*)d_in[1];
    const float* user_bias = (const float*)d_in[2];
    const float* item_bias = (const float*)d_in[3];
    const int*   edge      = (const int*)d_in[4];   // [2,E] flat
    const int*   user_ids  = (const int*)d_in[5];
    const int*   item_ids  = (const int*)d_in[6];
    float*       out       = (float*)d_out;

    const int E  = in_sizes[4] / 2;
    const int* eu = edge;
    const int* ei = edge + E;

    float* ws = (float*)d_ws;
    const size_t nodeF = (size_t)N_NODES * EMB_DIM;      // 9.6M floats
    float* deg = ws;                                     // 150016 pad
    float* xA  = ws + 150016;
    float* xB  = xA + nodeF;
    float* accb = xB + nodeF;

    const int B = 256;
    zero_f32  <<<(N_NODES + B - 1) / B, B, 0, stream>>>(deg, (size_t)N_NODES);
    deg_kernel<<<(E + B - 1) / B,       B, 0, stream>>>(eu, ei, deg, E);
    dinv_kernel<<<(N_NODES + B - 1) / B, B, 0, stream>>>(deg, N_NODES);
    init_emb  <<<(int)((nodeF + B - 1) / B), B, 0, stream>>>(user_emb, item_emb, xA, accb);

    float* xcur = xA;
    float* xnext = xB;
    for (int l = 0; l < N_LAYERS; ++l) {
        zero_f32<<<(int)((nodeF + B - 1) / B), B, 0, stream>>>(xnext, nodeF);
        long long threads = (long long)E * 32;
        scatter_kernel<<<(int)((threads + B - 1) / B), B, 0, stream>>>(
            eu, ei, deg, xcur, xnext, E);
        acc_add<<<(int)((nodeF + B - 1) / B), B, 0, stream>>>(accb, xnext, nodeF);
        float* t = xcur; xcur = xnext; xnext = t;
    }

    // 1024 tiles of 16 pairs, 8 waves/block -> 128 blocks
    final_dot_wmma<<<(BATCH / 16) / 8, 256, 0, stream>>>(
        accb, user_ids, item_ids, user_bias, item_bias, out);
}